// RelativePositionalEncoding_10041633538682
// MI455X (gfx1250) — compile-verified
//
#include <hip/hip_runtime.h>

// out[s,b,d] = x[s,b,d] + ((S-1)/2 - s)
// Pure streaming kernel: 256 MiB of HBM traffic, ~11.5 us floor at 23.3 TB/s.
// Strategy: B128 loads/stores with non-temporal hints, wave-uniform (SALU) bias.

typedef __attribute__((ext_vector_type(4))) float v4f;

namespace {
constexpr int SEQ  = 4096;          // seq_len
constexpr int ROWF = 4 * 2048;      // floats per s-index (batch * d_model) = 8192
constexpr int BLOCK_THREADS    = 256;
constexpr int FLOATS_PER_BLOCK = BLOCK_THREADS * 8;       // 2048 (2 x float4 per thread)
constexpr int BLOCKS_PER_ROW   = ROWF / FLOATS_PER_BLOCK; // 4  -> s = bid >> 2 (scalar)
}

__global__ __launch_bounds__(BLOCK_THREADS)
void rpe_add_bias_kernel(const float* __restrict__ x,
                         float* __restrict__ out,
                         float bias0 /* = (SEQ-1)/2 */) {
    const unsigned bid = blockIdx.x;

    // s is uniform across the whole block -> computed in SALU, bias lives in an SGPR.
    const unsigned s   = bid / BLOCKS_PER_ROW;
    const float bias   = bias0 - (float)s;

    const size_t base  = (size_t)bid * FLOATS_PER_BLOCK + (size_t)threadIdx.x * 4u;
    const size_t base1 = base + (size_t)BLOCK_THREADS * 4u;

    // Streamed once, no reuse: non-temporal B128 load/store (TH=NT) to avoid
    // polluting WGP$/L2 (working set 256 MiB > 192 MB L2).
    v4f a0 = __builtin_nontemporal_load((const v4f*)(x + base));
    v4f a1 = __builtin_nontemporal_load((const v4f*)(x + base1));

    a0 += bias;
    a1 += bias;

    __builtin_nontemporal_store(a0, (v4f*)(out + base));
    __builtin_nontemporal_store(a1, (v4f*)(out + base1));
}

extern "C" void kernel_launch(void* const* d_in, const int* in_sizes, int n_in,
                              void* d_out, int out_size, void* d_ws, size_t ws_size,
                              hipStream_t stream) {
    (void)in_sizes; (void)n_in; (void)d_ws; (void)ws_size;

    const float* x = (const float*)d_in[0];
    float* out     = (float*)d_out;

    // out_size = SEQ * 4 * 2048 = 33,554,432; exactly divisible by 2048.
    const int grid = out_size / FLOATS_PER_BLOCK;   // 16384 blocks
    const float bias0 = 0.5f * (float)(SEQ - 1);    // 2047.5

    rpe_add_bias_kernel<<<grid, BLOCK_THREADS, 0, stream>>>(x, out, bias0);
}